// EGRETLayer_71451075936281
// MI455X (gfx1250) — compile-verified
//
#include <hip/hip_runtime.h>
#include <math.h>

#define N_NODES 100000
#define N_EDGES 1600000
#define D_IN    128
#define D_OUT   64
#define NEG_SLOPE 0.01f

typedef __attribute__((ext_vector_type(2))) float v2f;
typedef __attribute__((ext_vector_type(8))) float v8f;

// ---- order-preserving float <-> uint key for atomic segment-max ----
__device__ __forceinline__ unsigned fkey(float f) {
    unsigned u = __float_as_uint(f);
    return (u & 0x80000000u) ? ~u : (u | 0x80000000u);
}
__device__ __forceinline__ float fdec(unsigned k) {
    unsigned u = (k & 0x80000000u) ? (k & 0x7fffffffu) : ~k;
    return __uint_as_float(u);
}

// ---------------- K0: init accumulators ----------------
__global__ void k_init(float* __restrict__ denom, unsigned* __restrict__ mkey,
                       float* __restrict__ hout) {
    int i = blockIdx.x * blockDim.x + threadIdx.x;
    if (i < N_NODES) { denom[i] = 0.0f; mkey[i] = 0u; }
    if (i < N_NODES * D_OUT) hout[i] = 0.0f;
}

// ---------------- K1: z = h @ W_fc.T via V_WMMA_F32_16X16X4_F32 ----------------
// One wave -> 16 rows x 64 cols of z. 4 accumulators (N-blocks), K looped by 4.
// A (16x4 f32): lane L<16 holds h[rb+L][k..k+1], lane L+16 holds h[rb+L][k+2..k+3].
// B (4x16 f32): B[k][n] = W_fc[n][k]; lane L<16 holds W_fc[nb*16+L][k..k+1],
//               lane L+16 holds W_fc[nb*16+L][k+2..k+3].
__global__ void k_gemm(const float* __restrict__ h, const float* __restrict__ Wfc,
                       float* __restrict__ z) {
    const int wave = (blockIdx.x * blockDim.x + threadIdx.x) >> 5;
    const int lane = threadIdx.x & 31;
    const int ntiles = N_NODES / 16;            // 6250, exact
    if (wave >= ntiles) return;                  // wave-uniform: EXEC stays all-1s
    const int rb = wave * 16;
    const int lm = lane & 15;
    const int kh = (lane >> 4) << 1;             // 0 or 2

    v8f c0 = {}, c1 = {}, c2 = {}, c3 = {};
    const float* arow = h   + (size_t)(rb + lm) * D_IN + kh;
    const float* br0  = Wfc + (size_t)( 0 + lm) * D_IN + kh;
    const float* br1  = Wfc + (size_t)(16 + lm) * D_IN + kh;
    const float* br2  = Wfc + (size_t)(32 + lm) * D_IN + kh;
    const float* br3  = Wfc + (size_t)(48 + lm) * D_IN + kh;

#pragma unroll 4
    for (int k = 0; k < D_IN; k += 4) {
        v2f a  = *(const v2f*)(arow + k);
        v2f b0 = *(const v2f*)(br0 + k);
        v2f b1 = *(const v2f*)(br1 + k);
        v2f b2 = *(const v2f*)(br2 + k);
        v2f b3 = *(const v2f*)(br3 + k);
        c0 = __builtin_amdgcn_wmma_f32_16x16x4_f32(false, a, false, b0, (short)0, c0, false, false);
        c1 = __builtin_amdgcn_wmma_f32_16x16x4_f32(false, a, false, b1, (short)0, c1, false, false);
        c2 = __builtin_amdgcn_wmma_f32_16x16x4_f32(false, a, false, b2, (short)0, c2, false, false);
        c3 = __builtin_amdgcn_wmma_f32_16x16x4_f32(false, a, false, b3, (short)0, c3, false, false);
    }

    // C layout: VGPR v, lanes 0-15 -> (M=v, N=lane); lanes 16-31 -> (M=v+8, N=lane-16)
    const int ch = (lane >> 4) * 8;
#pragma unroll
    for (int v = 0; v < 8; ++v) {
        float* zr = z + (size_t)(rb + ch + v) * D_OUT + lm;
        zr[0]  = c0[v];
        zr[16] = c1[v];
        zr[32] = c2[v];
        zr[48] = c3[v];
    }
}

// ---------------- K2: per-node attention scalars s1,s2 ----------------
__global__ void k_nodedot(const float* __restrict__ z, const float* __restrict__ Wattn,
                          float* __restrict__ s1, float* __restrict__ s2) {
    const int wave = (blockIdx.x * blockDim.x + threadIdx.x) >> 5;
    const int lane = threadIdx.x & 31;
    if (wave >= N_NODES) return;
    const float* zr = z + (size_t)wave * D_OUT;
    float z0 = zr[lane], z1 = zr[lane + 32];
    float p1 = z0 * Wattn[lane]      + z1 * Wattn[lane + 32];
    float p2 = z0 * Wattn[64 + lane] + z1 * Wattn[96 + lane];
#pragma unroll
    for (int off = 16; off; off >>= 1) {
        p1 += __shfl_xor(p1, off, 32);
        p2 += __shfl_xor(p2, off, 32);
    }
    if (lane == 0) { s1[wave] = p1; s2[wave] = p2; }
}

// ---------------- K3: edge logits + atomic segment max ----------------
__global__ void k_logits(const float* __restrict__ e, const int* __restrict__ src,
                         const int* __restrict__ dst,
                         const float* __restrict__ s1, const float* __restrict__ s2,
                         const float* __restrict__ Wedge, const float* __restrict__ Wattn,
                         float* __restrict__ a, unsigned* __restrict__ mkey) {
    int i = blockIdx.x * blockDim.x + threadIdx.x;
    if (i >= N_EDGES) return;
    float e0 = e[2 * i], e1 = e[2 * i + 1];
    float ex0 = e0 * Wedge[0] + e1 * Wedge[1];   // W_edge[0][:]
    float ex1 = e0 * Wedge[2] + e1 * Wedge[3];   // W_edge[1][:]
    int d = dst[i];
    float v = s1[src[i]] + s2[d] + ex0 * Wattn[2 * D_OUT] + ex1 * Wattn[2 * D_OUT + 1];
    v = v > 0.0f ? v : NEG_SLOPE * v;
    a[i] = v;
    atomicMax(mkey + d, fkey(v));
}

// ---------------- K4: exp + scatter numerator/denominator ----------------
// One wave per edge (x4 edges per wave); lanes cover dims {l, l+32}.
#define EDGES_PER_WAVE 4
__global__ void k_scatter(const float* __restrict__ e, const int* __restrict__ src,
                          const int* __restrict__ dst, const float* __restrict__ a,
                          const unsigned* __restrict__ mkey, const float* __restrict__ z,
                          const float* __restrict__ Wedge, const float* __restrict__ We2n,
                          float* __restrict__ denom, float* __restrict__ hout) {
    const int lane = threadIdx.x & 31;
    const int wave = (blockIdx.x * blockDim.x + threadIdx.x) >> 5;
    // W_e2n rows for this lane's two output dims (tiny, L2-hot)
    const float wa0 = We2n[2 * lane],        wa1 = We2n[2 * lane + 1];
    const float wb0 = We2n[2 * (lane + 32)], wb1 = We2n[2 * (lane + 32) + 1];
    const float we00 = Wedge[0], we01 = Wedge[1], we10 = Wedge[2], we11 = Wedge[3];

    int ebase = wave * EDGES_PER_WAVE;
#pragma unroll
    for (int t = 0; t < EDGES_PER_WAVE; ++t) {
        int i = ebase + t;
        if (i >= N_EDGES) break;
        int s = src[i], d = dst[i];
        float p = __expf(a[i] - fdec(mkey[d]));
        float e0 = e[2 * i], e1 = e[2 * i + 1];
        float ex0 = e0 * we00 + e1 * we01;
        float ex1 = e0 * we10 + e1 * we11;
        const float* zs = z + (size_t)s * D_OUT;          // coalesced 128B per half
        float m0 = p * (zs[lane]      + ex0 * wa0 + ex1 * wa1);
        float m1 = p * (zs[lane + 32] + ex0 * wb0 + ex1 * wb1);
        float* hr = hout + (size_t)d * D_OUT;
        atomicAdd(hr + lane,      m0);
        atomicAdd(hr + lane + 32, m1);
        if (lane == 0) atomicAdd(denom + d, p);
    }
}

// ---------------- K5: per-node normalize ----------------
__global__ void k_norm(float* __restrict__ hout, const float* __restrict__ denom) {
    int i = blockIdx.x * blockDim.x + threadIdx.x;
    if (i >= N_NODES * D_OUT) return;
    float dn = denom[i >> 6];
    hout[i] = dn > 0.0f ? hout[i] / dn : 0.0f;   // isolated nodes -> 0 (matches ref)
}

extern "C" void kernel_launch(void* const* d_in, const int* in_sizes, int n_in,
                              void* d_out, int out_size, void* d_ws, size_t ws_size,
                              hipStream_t stream) {
    const float* h     = (const float*)d_in[0];
    const float* e     = (const float*)d_in[1];
    const int*   src   = (const int*)  d_in[2];
    const int*   dst   = (const int*)  d_in[3];
    const float* Wfc   = (const float*)d_in[4];
    const float* Wattn = (const float*)d_in[5];
    const float* Wedge = (const float*)d_in[6];
    const float* We2n  = (const float*)d_in[7];
    float* hout = (float*)d_out;

    // workspace layout (floats)
    float*    z     = (float*)d_ws;                        // N*64
    float*    s1    = z  + (size_t)N_NODES * D_OUT;        // N
    float*    s2    = s1 + N_NODES;                        // N
    float*    a     = s2 + N_NODES;                        // E
    float*    denom = a  + N_EDGES;                        // N
    unsigned* mkey  = (unsigned*)(denom + N_NODES);        // N

    // K0: init
    k_init<<<(N_NODES * D_OUT + 255) / 256, 256, 0, stream>>>(denom, mkey, hout);

    // K1: WMMA GEMM (6250 tiles, 8 waves/block)
    k_gemm<<<(N_NODES / 16 + 7) / 8, 256, 0, stream>>>(h, Wfc, z);

    // K2: per-node dots (1 wave/node)
    k_nodedot<<<(N_NODES + 7) / 8, 256, 0, stream>>>(z, Wattn, s1, s2);

    // K3: edge logits + atomic max
    k_logits<<<(N_EDGES + 255) / 256, 256, 0, stream>>>(e, src, dst, s1, s2, Wedge, Wattn, a, mkey);

    // K4: exp + scatter (1 wave per 4 edges)
    {
        int waves = (N_EDGES + EDGES_PER_WAVE - 1) / EDGES_PER_WAVE;
        k_scatter<<<(waves + 7) / 8, 256, 0, stream>>>(e, src, dst, a, mkey, z,
                                                       Wedge, We2n, denom, hout);
    }

    // K5: normalize
    k_norm<<<(N_NODES * D_OUT + 255) / 256, 256, 0, stream>>>(hout, denom);
}